// EllipseLayer2_35459249996554
// MI455X (gfx1250) — compile-verified
//
#include <hip/hip_runtime.h>
#include <hip/hip_bf16.h>

typedef float v2f __attribute__((ext_vector_type(2)));
typedef float v8f __attribute__((ext_vector_type(8)));

namespace {
constexpr int kN      = 1024;          // points per (b, channel)
constexpr int kCh     = 64;            // channels (innermost dim of V/C)
constexpr int kThr    = 256;           // 8 waves (wave32)
constexpr int kSlices = 16;            // n-slices covered by a block
constexpr int kIters  = kN / kSlices;  // 64 iterations per thread
}

// One block per batch element b.
// Thread (g = tid&15, s = tid>>4) accumulates 8 fp32 moments for channels
// 4g..4g+3 over points n = s, s+16, s+32, ... (fully coalesced float4 loads:
// 16 consecutive threads read one contiguous 256B row of 64 channels).
//
// Closed form:  sum_n loss^2 = N - f^T M^{-1} f   with
//   M = [[S40,S31,S22],[S31,S22,S13],[S22,S13,S04]],  f = -[S20,S11,S02]
// so no second pass over the 128 MB of input is needed.
__global__ __launch_bounds__(kThr)
void ellipse_moments_kernel(const float* __restrict__ V,
                            const float* __restrict__ C,
                            float* __restrict__ out)
{
    // moment-major LDS staging, padded (68) to keep b128 stores ~conflict-free
    __shared__ float mom[8][kSlices][kCh + 4];
    __shared__ float vals[kCh];

    const int tid = threadIdx.x;
    const int g   = tid & 15;   // channel group -> channels 4g..4g+3
    const int s   = tid >> 4;   // n-slice 0..15
    const int b   = blockIdx.x;

    float acc[4][8];
    #pragma unroll
    for (int j = 0; j < 4; ++j)
        #pragma unroll
        for (int m = 0; m < 8; ++m) acc[j][m] = 0.0f;

    const size_t base = (size_t)b * kN * kCh + (size_t)s * kCh + 4 * g;
    const float* Vp = V + base;
    const float* Cp = C + base;

    #pragma unroll 4
    for (int i = 0; i < kIters; ++i) {
        const size_t off = (size_t)i * kSlices * kCh;
        float4 v4 = *(const float4*)(Vp + off);   // global_load_b128
        float4 c4 = *(const float4*)(Cp + off);   // global_load_b128
        float vv[4] = {v4.x, v4.y, v4.z, v4.w};
        float cc[4] = {c4.x, c4.y, c4.z, c4.w};
        #pragma unroll
        for (int j = 0; j < 4; ++j) {
            float v = vv[j], c = cc[j];
            float v2 = v * v, vc = v * c, c2 = c * c;
            acc[j][0] += v2;                       // S20
            acc[j][1] += vc;                       // S11
            acc[j][2] += c2;                       // S02
            acc[j][3] = fmaf(v2, v2, acc[j][3]);   // S40
            acc[j][4] = fmaf(v2, vc, acc[j][4]);   // S31
            acc[j][5] = fmaf(v2, c2, acc[j][5]);   // S22
            acc[j][6] = fmaf(vc, c2, acc[j][6]);   // S13
            acc[j][7] = fmaf(c2, c2, acc[j][7]);   // S04
        }
    }

    // stage partial moments: one float4 (4 channels) per moment
    #pragma unroll
    for (int m = 0; m < 8; ++m) {
        float4 w = make_float4(acc[0][m], acc[1][m], acc[2][m], acc[3][m]);
        *(float4*)&mom[m][s][4 * g] = w;           // ds_store_b128
    }
    __syncthreads();

    // one thread per channel: combine 16 slices, closed-form solve
    if (tid < kCh) {
        float S[8];
        #pragma unroll
        for (int m = 0; m < 8; ++m) {
            float t = 0.0f;
            #pragma unroll
            for (int sl = 0; sl < kSlices; ++sl) t += mom[m][sl][tid];
            S[m] = t;
        }
        const float f0 = -S[0], f1 = -S[1], f2 = -S[2];
        const float a = S[3], bb = S[4], c = S[5], d = S[6], e = S[7];
        const float adj00 = c * e - d * d;
        const float adj01 = c * d - bb * e;
        const float adj02 = bb * d - c * c;
        const float adj11 = a * e - c * c;
        const float adj12 = bb * c - a * d;
        const float adj22 = a * c - bb * bb;
        const float det = a * adj00 + bb * adj01 + c * adj02;
        const float q = f0 * f0 * adj00 + f1 * f1 * adj11 + f2 * f2 * adj22 +
                        2.0f * (f0 * f1 * adj01 + f0 * f2 * adj02 + f1 * f2 * adj12);
        vals[tid] = (float)kN - q / det;
    }
    __syncthreads();

    // Final 64 -> 1 reduction via fp32 WMMA (wave 0 only; scalar branch keeps
    // EXEC all-ones as WMMA requires). A(16x4) holds the 64 channel values,
    // B = ones, so D[m,n] = vals[4m]+...+vals[4m+3] for every column n.
    if (__builtin_amdgcn_readfirstlane(tid) == 0) {
        const int lane = tid;                       // 0..31
        const int m    = lane & 15;
        const int koff = (lane >> 4) << 1;          // lanes 0-15: K=0,1 ; 16-31: K=2,3
        v2f A;  A[0] = vals[4 * m + koff];  A[1] = vals[4 * m + koff + 1];
        v2f Bm; Bm[0] = 1.0f;               Bm[1] = 1.0f;
        v8f Cz = {0.f, 0.f, 0.f, 0.f, 0.f, 0.f, 0.f, 0.f};
        v8f D = __builtin_amdgcn_wmma_f32_16x16x4_f32(
            /*neg_a=*/false, A, /*neg_b=*/false, Bm,
            /*c_mod=*/(short)0, Cz, /*reuse_a=*/false, /*reuse_b=*/false);
        // lane 0 holds rows 0..7 of column 0, lane 16 holds rows 8..15
        float rs = D[0] + D[1] + D[2] + D[3] + D[4] + D[5] + D[6] + D[7];
        float tot = rs + __shfl_xor(rs, 16, 32);
        if (lane == 0)
            out[b] = tot * (1.0f / ((float)kCh * (float)kN));
    }
}

extern "C" void kernel_launch(void* const* d_in, const int* in_sizes, int n_in,
                              void* d_out, int out_size, void* d_ws, size_t ws_size,
                              hipStream_t stream) {
    (void)in_sizes; (void)n_in; (void)d_ws; (void)ws_size;
    const float* V = (const float*)d_in[0];
    const float* C = (const float*)d_in[1];
    float* out = (float*)d_out;
    // out_size == B == 256; one block per batch element
    ellipse_moments_kernel<<<out_size, kThr, 0, stream>>>(V, C, out);
}